// Model_18622978195581
// MI455X (gfx1250) — compile-verified
//
#include <hip/hip_runtime.h>
#include <hip/hip_bf16.h>

typedef __attribute__((ext_vector_type(2))) float v2f;
typedef __attribute__((ext_vector_type(8))) float v8f;
typedef __attribute__((ext_vector_type(4))) int   v4i;

typedef __attribute__((address_space(1))) v4i* gbl_v4i_p;
typedef __attribute__((address_space(3))) v4i* lds_v4i_p;

#if defined(__has_builtin)
#if __has_builtin(__builtin_amdgcn_global_load_async_to_lds_b128)
#define HAVE_ASYNC_LDS 1
#endif
#endif
#ifndef HAVE_ASYNC_LDS
#define HAVE_ASYNC_LDS 0
#endif

__device__ __forceinline__ void wait_asynccnt0() {
#if defined(__has_builtin) && __has_builtin(__builtin_amdgcn_s_wait_asynccnt)
    __builtin_amdgcn_s_wait_asynccnt(0);
#else
    asm volatile("s_wait_asynccnt 0x0" ::: "memory");
#endif
}

// ---------------------------------------------------------------------------
// Degree / normalization precompute
// ---------------------------------------------------------------------------
__global__ void k_degree(const long long* __restrict__ dst, float* __restrict__ deg, int E) {
    int i = blockIdx.x * blockDim.x + threadIdx.x;
    if (i < E) atomicAdd(&deg[(int)dst[i]], 1.0f);
}

__global__ void k_dinv(float* __restrict__ deg, int N) {
    int i = blockIdx.x * blockDim.x + threadIdx.x;
    if (i < N) deg[i] = rsqrtf(deg[i] + 1.0f);   // +1 for the self-loop; deg >= 1 always
}

// ---------------------------------------------------------------------------
// WMMA fp32 GEMM: C[M x Ncols] = act(A [+ bias]) * B
//   A row-major lda=K; B row-major ldb=Ncols; M%16==0; Ncols % (64*CT) == 0.
//   block = 128 threads (4 waves); block tile = 16 rows x (64*CT) cols;
//   each wave owns CT adjacent 16x16 column tiles so the A fragment (one
//   ds_load_b64 per k-step) feeds CT v_wmma_f32_16x16x4_f32 issues.
//   A tile staged in LDS with +4 float pad (stride 132/260 -> conflict-free
//   v2f fragment reads across all 64 banks). Layer-1 path stages A with
//   global_load_async_to_lds_b128 (ASYNCcnt); the fused bias+relu path
//   stages manually since the activation must be applied in flight.
// ---------------------------------------------------------------------------
template<int K, int CT, bool PRE_BIAS_RELU>
__global__ __launch_bounds__(128) void k_gemm(const float* __restrict__ A,
                                              const float* __restrict__ B,
                                              const float* __restrict__ bias,
                                              float* __restrict__ C,
                                              int Ncols) {
    constexpr int LDA = K + 4;
    __shared__ float As[16 * LDA];

    const int tid  = threadIdx.x;
    const int row0 = blockIdx.x * 16;
    const int col0 = blockIdx.y * (64 * CT) + (tid >> 5) * (16 * CT);

    // ---- stage A tile (16 x K) into LDS ----
#if HAVE_ASYNC_LDS
    if constexpr (!PRE_BIAS_RELU) {
        for (int idx = tid; idx < 16 * (K / 4); idx += 128) {
            const int r = idx / (K / 4);
            const int c = (idx % (K / 4)) * 4;
            __builtin_amdgcn_global_load_async_to_lds_b128(
                (gbl_v4i_p)&A[(size_t)(row0 + r) * K + c],
                (lds_v4i_p)&As[r * LDA + c],
                /*imm offset=*/0, /*cpol=*/0);
        }
        wait_asynccnt0();
    } else
#endif
    {
        for (int idx = tid; idx < 16 * (K / 4); idx += 128) {
            const int r = idx / (K / 4);
            const int c = (idx % (K / 4)) * 4;
            float4 v = *(const float4*)&A[(size_t)(row0 + r) * K + c];
            if constexpr (PRE_BIAS_RELU) {
                v.x = fmaxf(v.x + bias[c + 0], 0.0f);
                v.y = fmaxf(v.y + bias[c + 1], 0.0f);
                v.z = fmaxf(v.z + bias[c + 2], 0.0f);
                v.w = fmaxf(v.w + bias[c + 3], 0.0f);
            }
            *(float4*)&As[r * LDA + c] = v;   // 16B aligned: LDA % 4 == 0
        }
    }
    __syncthreads();

    const int lane = tid & 31;
    const int n    = lane & 15;       // tile column / A row
    const int half = lane >> 4;       // K sub-slot select

    v8f acc[CT];
#pragma unroll
    for (int t = 0; t < CT; ++t) acc[t] = (v8f){};

    for (int k = 0; k < K; k += 4) {
        const int k0 = k + half * 2;
        const v2f a = *(const v2f*)&As[n * LDA + k0];          // ds_load_b64, reused CT times
#pragma unroll
        for (int t = 0; t < CT; ++t) {
            v2f b;
            b.x = B[(size_t)(k0 + 0) * Ncols + col0 + t * 16 + n];
            b.y = B[(size_t)(k0 + 1) * Ncols + col0 + t * 16 + n];
            acc[t] = __builtin_amdgcn_wmma_f32_16x16x4_f32(
                         false, a, false, b, (short)0, acc[t], false, false);
        }
    }

#pragma unroll
    for (int t = 0; t < CT; ++t) {
#pragma unroll
        for (int r = 0; r < 8; ++r) {
            const int m = r + half * 8;
            C[(size_t)(row0 + m) * Ncols + col0 + t * 16 + n] = acc[t][r];
        }
    }
}

// ---------------------------------------------------------------------------
// Edge scatter-add: out[dst] += H[src] * dinv[src]*dinv[dst]
// One wave per message; lanes carry float4 chunks of the F features.
// Virtual messages e in [E, E+N) are the self-loops (src = dst = e - E).
// Wave id forced uniform so edge-index / dinv loads become scalar (SMEM).
// ---------------------------------------------------------------------------
template<int F>
__global__ void k_agg(const float* __restrict__ H,
                      const long long* __restrict__ src,
                      const long long* __restrict__ dst,
                      const float* __restrict__ dinv,
                      float* __restrict__ out, int E, int N) {
    const int wave = __builtin_amdgcn_readfirstlane(
        (int)((((long long)blockIdx.x * blockDim.x) + threadIdx.x) >> 5));
    const int lane = threadIdx.x & 31;
    if (wave >= E + N) return;

    int s, d;
    if (wave < E) { s = (int)src[wave]; d = (int)dst[wave]; }
    else          { s = d = wave - E; }

    const float nrm = dinv[s] * dinv[d];
    const float* hs = H   + (size_t)s * F;
    float*       od = out + (size_t)d * F;

#pragma unroll
    for (int it = 0; it < F / 128; ++it) {
        const int c = lane * 4 + it * 128;
        const float4 v = *(const float4*)&hs[c];
        atomicAdd(&od[c + 0], v.x * nrm);
        atomicAdd(&od[c + 1], v.y * nrm);
        atomicAdd(&od[c + 2], v.z * nrm);
        atomicAdd(&od[c + 3], v.w * nrm);
    }
}

// ---------------------------------------------------------------------------
// Final epilogue: out = relu(out + bias), bias indexed by (i & mask)
// ---------------------------------------------------------------------------
__global__ void k_bias_relu(float* __restrict__ out, const float* __restrict__ bias,
                            int total, int mask) {
    int i = blockIdx.x * blockDim.x + threadIdx.x;
    if (i < total) out[i] = fmaxf(out[i] + bias[i & mask], 0.0f);
}

// ---------------------------------------------------------------------------
// Launch: x, edge_index(int64 [2,E]), W1[128,256], b1, W2[256,128], b2
// ---------------------------------------------------------------------------
extern "C" void kernel_launch(void* const* d_in, const int* in_sizes, int n_in,
                              void* d_out, int out_size, void* d_ws, size_t ws_size,
                              hipStream_t stream) {
    const float*      x   = (const float*)d_in[0];
    const long long*  ei  = (const long long*)d_in[1];
    const float*      W1  = (const float*)d_in[2];
    const float*      b1  = (const float*)d_in[3];
    const float*      W2  = (const float*)d_in[4];
    const float*      b2  = (const float*)d_in[5];
    float*            out = (float*)d_out;

    const int N = in_sizes[0] / 128;       // 100000
    const int E = in_sizes[1] / 2;         // 600000
    const long long* esrc = ei;            // edge_index[0]
    const long long* edst = ei + E;        // edge_index[1]

    // Workspace layout (floats): dinv[N] | H1[N*256] | A1[N*256] | H2[N*128]
    float* ws   = (float*)d_ws;
    float* dinv = ws;
    float* H1   = dinv + N;
    float* A1   = H1 + (size_t)N * 256;
    float* H2   = A1 + (size_t)N * 256;

    // --- normalization ---
    (void)hipMemsetAsync(dinv, 0, (size_t)N * sizeof(float), stream);
    k_degree<<<(E + 255) / 256, 256, 0, stream>>>(edst, dinv, E);
    k_dinv<<<(N + 255) / 256, 256, 0, stream>>>(dinv, N);

    // --- layer 1: H1 = x @ W1 ; A1 = scatter(H1) ---
    {
        dim3 g(N / 16, 1);                              // 4 waves x 4 tiles = 256 cols
        k_gemm<128, 4, false><<<g, 128, 0, stream>>>(x, W1, nullptr, H1, 256);
    }
    (void)hipMemsetAsync(A1, 0, (size_t)N * 256 * sizeof(float), stream);
    {
        const long long th = (long long)(E + N) * 32;
        k_agg<256><<<(int)((th + 255) / 256), 256, 0, stream>>>(H1, esrc, edst, dinv, A1, E, N);
    }

    // --- layer 2: H2 = relu(A1 + b1) @ W2 (bias+relu fused into A staging) ---
    {
        dim3 g(N / 16, 1);                              // 4 waves x 2 tiles = 128 cols
        k_gemm<256, 2, true><<<g, 128, 0, stream>>>(A1, W2, b1, H2, 128);
    }
    (void)hipMemsetAsync(out, 0, (size_t)N * 128 * sizeof(float), stream);
    {
        const long long th = (long long)(E + N) * 32;
        k_agg<128><<<(int)((th + 255) / 256), 256, 0, stream>>>(H2, esrc, edst, dinv, out, E, N);
    }

    // --- final bias + relu in place on d_out ---
    k_bias_relu<<<(N * 128 + 255) / 256, 256, 0, stream>>>(out, b2, N * 128, 127);
}